// LegoGNNEncoder_65481071395096
// MI455X (gfx1250) — compile-verified
//
#include <hip/hip_runtime.h>
#include <hip/hip_bf16.h>

typedef __attribute__((ext_vector_type(2))) float v2f;
typedef __attribute__((ext_vector_type(8))) float v8f;

// ---------------------------------------------------------------------------
// Degree / normalization
// ---------------------------------------------------------------------------
__global__ void k_deg_init(float* __restrict__ deg, int n) {
    int i = blockIdx.x * blockDim.x + threadIdx.x;
    if (i < n) deg[i] = 1.0f;                       // self-loop contribution
}

__global__ void k_deg_edges(const int* __restrict__ dst, float* __restrict__ deg, int e) {
    int i = blockIdx.x * blockDim.x + threadIdx.x;
    if (i < e) unsafeAtomicAdd(&deg[dst[i]], 1.0f); // global_atomic_add_f32
}

__global__ void k_dinv(const float* __restrict__ deg, float* __restrict__ dinv, int n) {
    int i = blockIdx.x * blockDim.x + threadIdx.x;
    if (i < n) dinv[i] = __frsqrt_rn(deg[i]);       // deg >= 1 always
}

// ---------------------------------------------------------------------------
// Aggregation: agg[i] = dinv[i]^2 * feat[i]  +  sum_e dinv[s]*dinv[d]*feat[s]
// nodeShift = log2(C/4)
// ---------------------------------------------------------------------------
__global__ void k_agg_init(const float4* __restrict__ feat, const float* __restrict__ dinv,
                           float4* __restrict__ agg, int total4, int nodeShift) {
    int t = blockIdx.x * blockDim.x + threadIdx.x;
    if (t >= total4) return;
    int i = t >> nodeShift;
    float w = dinv[i]; w *= w;
    float4 v = feat[t];
    agg[t] = make_float4(v.x * w, v.y * w, v.z * w, v.w * w);
}

// tpeShift = log2(C/4); one thread handles one float4 of one edge
__global__ void k_agg_edges(const float4* __restrict__ feat, const int* __restrict__ src,
                            const int* __restrict__ dst, const float* __restrict__ dinv,
                            float* __restrict__ agg, int total, int tpeShift) {
    int t = blockIdx.x * blockDim.x + threadIdx.x;
    if (t >= total) return;
    int e  = t >> tpeShift;
    int c4 = t & ((1 << tpeShift) - 1);
    int Cq = 1 << tpeShift;
    int s = src[e], d = dst[e];
    float w = dinv[s] * dinv[d];
    float4 v = feat[(size_t)s * Cq + c4];
    float* ap = agg + (((size_t)d * Cq + c4) << 2);
    unsafeAtomicAdd(ap + 0, v.x * w);
    unsafeAtomicAdd(ap + 1, v.y * w);
    unsafeAtomicAdd(ap + 2, v.z * w);
    unsafeAtomicAdd(ap + 3, v.w * w);
}

// ---------------------------------------------------------------------------
// GEMM 1 (WMMA f32 16x16x4): H[nrows x 128] = relu(A[nrows x 64] @ W[64 x 128] + b)
// One wave = 16 rows x 128 cols (8 column tiles).
// ---------------------------------------------------------------------------
__global__ void k_gemm1(const float* __restrict__ A, const float* __restrict__ W,
                        const float* __restrict__ bias, float* __restrict__ H, int nrows) {
    const int wave = blockIdx.x * (blockDim.x >> 5) + (threadIdx.x >> 5);
    const int lane = threadIdx.x & 31;
    const int row0 = wave << 4;
    if (row0 >= nrows) return;              // wave-uniform: EXEC stays all-1s
    const int m  = lane & 15;
    const int kh = lane >> 4;

    v8f acc[8];
    #pragma unroll
    for (int t = 0; t < 8; ++t) acc[t] = (v8f){0.f,0.f,0.f,0.f,0.f,0.f,0.f,0.f};

    const float* arow = A + (size_t)(row0 + m) * 64;

    for (int k0 = 0; k0 < 64; k0 += 4) {
        v2f a = *(const v2f*)(arow + k0 + 2 * kh);      // K = k0+2kh, k0+2kh+1
        #pragma unroll
        for (int t = 0; t < 8; ++t) {
            const float* bp = W + (size_t)(k0 + 2 * kh) * 128 + t * 16 + m;
            v2f b; b.x = bp[0]; b.y = bp[128];
            acc[t] = __builtin_amdgcn_wmma_f32_16x16x4_f32(
                false, a, false, b, (short)0, acc[t], false, false);
        }
    }

    #pragma unroll
    for (int t = 0; t < 8; ++t) {
        const int c = t * 16 + m;
        const float bv = bias[c];
        #pragma unroll
        for (int v = 0; v < 8; ++v) {
            const int r = row0 + v + 8 * kh;
            H[(size_t)r * 128 + c] = fmaxf(acc[t][v] + bv, 0.0f);
        }
    }
}

// ---------------------------------------------------------------------------
// GEMM 2 (WMMA f32 16x16x4): fused mu / logstd heads.
//   mu = A @ Wmu + bmu ;  ls = min(A @ Wls + bls, 10)
// Column tiles 0-3 -> Wmu, 4-7 -> Wls (both 128x64).
// ---------------------------------------------------------------------------
__global__ void k_gemm2(const float* __restrict__ A,
                        const float* __restrict__ Wmu, const float* __restrict__ bmu,
                        const float* __restrict__ Wls, const float* __restrict__ bls,
                        float* __restrict__ mu, float* __restrict__ ls, int nrows) {
    const int wave = blockIdx.x * (blockDim.x >> 5) + (threadIdx.x >> 5);
    const int lane = threadIdx.x & 31;
    const int row0 = wave << 4;
    if (row0 >= nrows) return;
    const int m  = lane & 15;
    const int kh = lane >> 4;

    v8f acc[8];
    #pragma unroll
    for (int t = 0; t < 8; ++t) acc[t] = (v8f){0.f,0.f,0.f,0.f,0.f,0.f,0.f,0.f};

    const float* arow = A + (size_t)(row0 + m) * 128;

    for (int k0 = 0; k0 < 128; k0 += 4) {
        v2f a = *(const v2f*)(arow + k0 + 2 * kh);
        #pragma unroll
        for (int t = 0; t < 8; ++t) {
            const float* W = (t < 4) ? Wmu : Wls;
            const float* bp = W + (size_t)(k0 + 2 * kh) * 64 + (t & 3) * 16 + m;
            v2f b; b.x = bp[0]; b.y = bp[64];
            acc[t] = __builtin_amdgcn_wmma_f32_16x16x4_f32(
                false, a, false, b, (short)0, acc[t], false, false);
        }
    }

    #pragma unroll
    for (int t = 0; t < 8; ++t) {
        const int c = (t & 3) * 16 + m;
        if (t < 4) {
            const float bv = bmu[c];
            #pragma unroll
            for (int v = 0; v < 8; ++v) {
                const int r = row0 + v + 8 * kh;
                mu[(size_t)r * 64 + c] = acc[t][v] + bv;
            }
        } else {
            const float bv = bls[c];
            #pragma unroll
            for (int v = 0; v < 8; ++v) {
                const int r = row0 + v + 8 * kh;
                ls[(size_t)r * 64 + c] = fminf(acc[t][v] + bv, 10.0f);
            }
        }
    }
}

// ---------------------------------------------------------------------------
// Launch
// ---------------------------------------------------------------------------
extern "C" void kernel_launch(void* const* d_in, const int* in_sizes, int n_in,
                              void* d_out, int out_size, void* d_ws, size_t ws_size,
                              hipStream_t stream) {
    const float* x   = (const float*)d_in[0];
    const int*   ei  = (const int*)  d_in[1];
    const float* W1  = (const float*)d_in[2];
    const float* b1  = (const float*)d_in[3];
    const float* Wmu = (const float*)d_in[4];
    const float* bmu = (const float*)d_in[5];
    const float* Wls = (const float*)d_in[6];
    const float* bls = (const float*)d_in[7];
    float* out = (float*)d_out;

    const int N = in_sizes[0] / 64;      // 100000
    const int E = in_sizes[1] / 2;       // 1000000
    const int* src = ei;
    const int* dst = ei + E;

    // workspace layout: deg[N] | dinv[N] | bufA[N*128] | bufH[N*128]
    float* deg  = (float*)d_ws;
    float* dinv = deg + N;
    float* bufA = dinv + N;
    float* bufH = bufA + (size_t)N * 128;

    const int B = 256;
    auto cdiv = [](long long a, long long b) { return (int)((a + b - 1) / b); };

    // 1) degree + normalization
    k_deg_init <<<cdiv(N, B), B, 0, stream>>>(deg, N);
    k_deg_edges<<<cdiv(E, B), B, 0, stream>>>(dst, deg, E);
    k_dinv     <<<cdiv(N, B), B, 0, stream>>>(deg, dinv, N);

    // 2) layer-1 aggregation of raw features (64 ch):  agg0 = D^-1/2 Ahat D^-1/2 x
    k_agg_init <<<cdiv((long long)N * 16, B), B, 0, stream>>>((const float4*)x, dinv,
                                                             (float4*)bufA, N * 16, 4);
    k_agg_edges<<<cdiv((long long)E * 16, B), B, 0, stream>>>((const float4*)x, src, dst,
                                                             dinv, bufA, E * 16, 4);

    // 3) h = relu(agg0 @ W1 + b1)   (WMMA f32)
    const int gemmBlocks = cdiv((long long)cdiv(N, 16), 8);
    k_gemm1<<<gemmBlocks, 256, 0, stream>>>(bufA, W1, b1, bufH, N);

    // 4) layer-2 aggregation of h (128 ch): agg1 = D^-1/2 Ahat D^-1/2 h
    k_agg_init <<<cdiv((long long)N * 32, B), B, 0, stream>>>((const float4*)bufH, dinv,
                                                             (float4*)bufA, N * 32, 5);
    k_agg_edges<<<cdiv((long long)E * 32, B), B, 0, stream>>>((const float4*)bufH, src, dst,
                                                             dinv, bufA, E * 32, 5);

    // 5) mu / logstd heads in one fused WMMA GEMM
    k_gemm2<<<gemmBlocks, 256, 0, stream>>>(bufA, Wmu, bmu, Wls, bls,
                                            out, out + (size_t)N * 64, N);
}